// TemporalDeform_76785425318168
// MI455X (gfx1250) — compile-verified
//
#include <hip/hip_runtime.h>
#include <math.h>

// ---- problem constants (fixed by the reference) ----
#define NT   64
#define TT   8
#define NB   8          // N = NT / T
#define CC   256
#define HW   3136       // 56*56
#define HW4  784        // HW / 4
#define KKD  768        // C * 3  (conv K-dim)

typedef float v2f __attribute__((ext_vector_type(2)));
typedef float v8f __attribute__((ext_vector_type(8)));
typedef int   v4i __attribute__((ext_vector_type(4)));
typedef __attribute__((address_space(1))) v4i gv4i;   // global int4
typedef __attribute__((address_space(3))) v4i lv4i;   // LDS int4

__device__ __forceinline__ float sigmoid_(float x) { return 1.0f / (1.0f + expf(-x)); }

// ---------------------------------------------------------------------------
// Kernel 1: spatial mean pooling. One block per (nt, c) plane; raw sums out.
// ---------------------------------------------------------------------------
__global__ void td_pool_kernel(const float* __restrict__ x, float* __restrict__ P) {
    __shared__ float sdata[128];
    const int plane = blockIdx.x;                      // nt*C + c
    const float4* xp = (const float4*)x + (size_t)plane * HW4;
    float acc = 0.0f;
    for (int i = threadIdx.x; i < HW4; i += 128) {
        float4 v = xp[i];
        acc += v.x + v.y + v.z + v.w;
    }
    sdata[threadIdx.x] = acc;
    __syncthreads();
    for (int s = 64; s > 0; s >>= 1) {
        if (threadIdx.x < s) sdata[threadIdx.x] += sdata[threadIdx.x + s];
        __syncthreads();
    }
    if (threadIdx.x == 0) P[plane] = sdata[0];
}

// ---------------------------------------------------------------------------
// Kernel 2: bias-net + weight-net as one f32 WMMA GEMM (3x768 @ 768x64).
// A (filters, pre-scaled by 1/HW) and B (pooled sums) are staged in LDS;
// the B panel is copied with gfx1250 async global->LDS B128 when available.
// 4 waves; wave w owns output columns [16w, 16w+16) = (n,t) pairs n=2w,2w+1.
// ---------------------------------------------------------------------------
__global__ void td_net_kernel(const float* __restrict__ P,    // raw sums (NT, C)
                              const float* __restrict__ cbw,  // (1,C,3) -> 768
                              const float* __restrict__ cbb,  // (1,)
                              const float* __restrict__ fcw,  // (8,8)
                              const float* __restrict__ fcb,  // (8,)
                              const float* __restrict__ lw,   // (2,8)
                              const float* __restrict__ lb,   // (2,)
                              const float* __restrict__ wwp,  // (2,C,3) -> 1536
                              const float* __restrict__ wcb,  // (2,)
                              int*   __restrict__ shift_i,    // (N,4)
                              float* __restrict__ w0f,        // (N,4)
                              float* __restrict__ wgt) {      // (N,T,2)
    __shared__ float ldsP[NT * CC];    // 16384 f = 64 KB : B panel (raw sums)
    __shared__ float ldsA[3 * KKD];    //  2304 f         : A panel (scaled)
    __shared__ float lds_b[64];        // bias-conv output [n][t]
    __shared__ float lds_w[128];       // 2*sigmoid(wconv) [n][t][j]

    const float inv = 1.0f / (float)HW;

    // ---- stage B panel: global -> LDS (async path on gfx1250) ----
#if defined(__gfx1250__) && __has_builtin(__builtin_amdgcn_global_load_async_to_lds_b128)
    for (int i = threadIdx.x; i < (NT * CC) / 4; i += 128) {
        __builtin_amdgcn_global_load_async_to_lds_b128(
            (gv4i*)(void*)(const_cast<float*>(P) + 4 * i),
            (lv4i*)(void*)(&ldsP[4 * i]),
            0, 0);
    }
#if __has_builtin(__builtin_amdgcn_s_wait_asynccnt)
    __builtin_amdgcn_s_wait_asynccnt(0);
#else
    asm volatile("s_wait_asynccnt 0x0" ::: "memory");
#endif
#else
    for (int i = threadIdx.x; i < (NT * CC) / 4; i += 128)
        ((float4*)ldsP)[i] = ((const float4*)P)[i];
#endif

    // ---- stage A panel (fold the 1/HW mean scale into the filters) ----
    for (int i = threadIdx.x; i < 3 * KKD; i += 128)
        ldsA[i] = (i < KKD ? cbw[i] : wwp[i - KKD]) * inv;

    __syncthreads();

    const int lane = threadIdx.x & 31;
    const int wave = threadIdx.x >> 5;
    const int half = lane >> 4;     // K offset selector (0 -> K, 1 -> K+2)
    const int l    = lane & 15;     // A row (M) and B column within tile
    const int col  = wave * 16 + l; // 0..63 == n*8 + t
    const int n = col >> 3, t = col & 7;
    const int arow = (l < 3 ? l : 0) * KKD;   // clamped row; zeroed below

    v8f acc = {};
    for (int s = 0; s < 192; ++s) {
        v2f a, b;
#pragma unroll
        for (int r = 0; r < 2; ++r) {
            const int kk = 4 * s + r + 2 * half;     // (c,k) flat index
            // A: unconditional LDS load + cndmask zero for rows >= 3
            const float av = ldsA[arow + kk];
            a[r] = (l < 3) ? av : 0.0f;
            // B: shifted pooled descriptor, zero-padded in t
            const int cc = kk / 3;
            const int k  = kk - 3 * cc;
            const int ts = t + k - 1;
            const int tc = ts < 0 ? 0 : (ts > 7 ? 7 : ts);
            const float bv = ldsP[(n * 8 + tc) * CC + cc];
            b[r] = (ts >= 0 && ts < 8) ? bv : 0.0f;
        }
        // D = A x B + C  (f32 16x16x4)
        acc = __builtin_amdgcn_wmma_f32_16x16x4_f32(false, a, false, b,
                                                    (short)0, acc, false, false);
    }

    // C/D layout: lanes 0-15 hold rows 0..7 in acc[0..7]; we need rows 0..2.
    if (half == 0) {
        lds_b[col]         = acc[0] + cbb[0];
        lds_w[col * 2 + 0] = 2.0f * sigmoid_(acc[1] + wcb[0]);
        lds_w[col * 2 + 1] = 2.0f * sigmoid_(acc[2] + wcb[1]);
    }
    __syncthreads();

    if (threadIdx.x < 8) {
        const int nn = threadIdx.x;
        float h[8];
#pragma unroll
        for (int i = 0; i < 8; ++i) {
            float s = fcb[i];
#pragma unroll
            for (int j = 0; j < 8; ++j) s += lds_b[nn * 8 + j] * fcw[i * 8 + j];
            h[i] = fmaxf(s, 0.0f);
        }
        float o0 = lb[0], o1 = lb[1];
#pragma unroll
        for (int j = 0; j < 8; ++j) { o0 += h[j] * lw[j]; o1 += h[j] * lw[8 + j]; }
        const float b0 = 4.0f * (sigmoid_(o0) - 0.5f);
        const float b1 = 4.0f * (sigmoid_(o1) - 0.5f);
        const float bias[4] = { b0, b1, -b0, -b1 };
#pragma unroll
        for (int g = 0; g < 4; ++g) {
            const float bf = floorf(bias[g]);
            shift_i[nn * 4 + g] = (int)bf;
            w0f[nn * 4 + g]     = 1.0f - (bias[g] - bf);
        }
    }
    if (threadIdx.x < 128) wgt[threadIdx.x] = lds_w[threadIdx.x];
}

// ---------------------------------------------------------------------------
// Kernel 3: temporal linear sampler + channel weighting. One thread per
// (n, c, hw4) column; rolling over t so each source plane is loaded once
// ((T+1) loads for T outputs). Fully coalesced B128 traffic.
// ---------------------------------------------------------------------------
__global__ void td_sampler_kernel(const float* __restrict__ x,
                                  const int*   __restrict__ shift_i,
                                  const float* __restrict__ w0f,
                                  const float* __restrict__ wgt,
                                  float* __restrict__ out) {
    const unsigned total = NB * CC * HW4;
    const unsigned i = blockIdx.x * blockDim.x + threadIdx.x;
    if (i >= total) return;
    const unsigned q  = i % HW4;
    const unsigned pc = i / HW4;      // n*C + c
    const unsigned c  = pc % CC;
    const unsigned n  = pc / CC;
    const unsigned g  = c >> 6;       // 4 channel groups of 64

    const int   sh = shift_i[n * 4 + g];
    const float w0 = w0f[n * 4 + g];
    const float w1 = 1.0f - w0;

    const float4* x4 = (const float4*)x;
    float4*       o4 = (float4*)out;
    const size_t tstr = (size_t)CC * HW4;                    // t stride (float4)
    const size_t base = ((size_t)(n * 8) * CC + c) * HW4 + q;

    auto loadT = [&](int tq) -> float4 {
        const int tc = tq < 0 ? 0 : (tq > 7 ? 7 : tq);
        float4 v = x4[base + (size_t)tc * tstr];
        const float m = (tq >= 0 && tq <= 7) ? 1.0f : 0.0f;
        v.x *= m; v.y *= m; v.z *= m; v.w *= m;
        return v;
    };

    float4 prev = loadT(sh);            // t=0 floor plane
#pragma unroll
    for (int t = 0; t < 8; ++t) {
        float4 cur = loadT(t + sh + 1); // ceil plane == next floor plane
        const float wc = wgt[(n * 8 + t) * 2 + (g & 1)];
        const float a0 = w0 * wc, a1 = w1 * wc;
        float4 r;
        r.x = a0 * prev.x + a1 * cur.x;
        r.y = a0 * prev.y + a1 * cur.y;
        r.z = a0 * prev.z + a1 * cur.z;
        r.w = a0 * prev.w + a1 * cur.w;
        o4[base + (size_t)t * tstr] = r;
        prev = cur;
    }
}

// ---------------------------------------------------------------------------
extern "C" void kernel_launch(void* const* d_in, const int* in_sizes, int n_in,
                              void* d_out, int out_size, void* d_ws, size_t ws_size,
                              hipStream_t stream) {
    const float* x   = (const float*)d_in[0];
    const float* cbw = (const float*)d_in[1];
    const float* cbb = (const float*)d_in[2];
    const float* fcw = (const float*)d_in[3];
    const float* fcb = (const float*)d_in[4];
    const float* lw  = (const float*)d_in[5];
    const float* lb  = (const float*)d_in[6];
    const float* wwp = (const float*)d_in[7];
    const float* wcb = (const float*)d_in[8];
    // d_in[9] = n_segment (hardcoded 8)

    float* P       = (float*)d_ws;              // NT*C raw sums (16384 f)
    int*   shift_i = (int*)(P + NT * CC);       // 32 ints
    float* w0f     = (float*)(shift_i + 32);    // 32 f
    float* wgt     = w0f + 32;                  // 128 f
    float* out     = (float*)d_out;

    td_pool_kernel<<<NT * CC, 128, 0, stream>>>(x, P);
    td_net_kernel<<<1, 128, 0, stream>>>(P, cbw, cbb, fcw, fcb, lw, lb, wwp, wcb,
                                         shift_i, w0f, wgt);
    const unsigned total = NB * CC * HW4;
    td_sampler_kernel<<<(total + 255) / 256, 256, 0, stream>>>(x, shift_i, w0f, wgt, out);
}